// AdjustableConvolution2d_44049184588028
// MI455X (gfx1250) — compile-verified
//
#include <hip/hip_runtime.h>
#include <hip/hip_bf16.h>

typedef __attribute__((ext_vector_type(2))) float v2f;
typedef __attribute__((ext_vector_type(8))) float v8f;

#define BS 8
#define C 256
#define H 64
#define W 64
#define SQ 32
#define KK 9
#define CKK (C * KK)   // 2304

// ---------------------------------------------------------------------------
// Kernel 1: per-sample filter generation.
// t = temp_feat@Wt + bt ; raw = (t@Wf + bf)/100 ; softmax over the 9 taps.
// One block per batch sample.
// ---------------------------------------------------------------------------
__global__ __launch_bounds__(256) void gen_filters(
    const float* __restrict__ temp_feat, const float* __restrict__ Wt,
    const float* __restrict__ bt, const float* __restrict__ Wf,
    const float* __restrict__ bf, float* __restrict__ filt_out) {
  __shared__ float tf[C];
  __shared__ float t[SQ];
  __shared__ float raw[CKK];

  const int b = blockIdx.x;
  const int tid = threadIdx.x;

  tf[tid] = temp_feat[b * C + tid];
  __syncthreads();

  if (tid < SQ) {
    float s = bt[tid];
    #pragma unroll 8
    for (int i = 0; i < C; ++i) s += tf[i] * Wt[i * SQ + tid];
    t[tid] = s;
  }
  __syncthreads();

  for (int j = tid; j < CKK; j += 256) {
    float s = bf[j];
    #pragma unroll
    for (int q = 0; q < SQ; ++q) s += t[q] * Wf[q * CKK + j];
    raw[j] = s * 0.01f;  // / SOFTMAX_TEMP
  }
  __syncthreads();

  // one thread per channel: softmax over 9 taps
  const int base = tid * KK;
  float m = raw[base];
  #pragma unroll
  for (int q = 1; q < KK; ++q) m = fmaxf(m, raw[base + q]);
  float e[KK];
  float sum = 0.f;
  #pragma unroll
  for (int q = 0; q < KK; ++q) { e[q] = __expf(raw[base + q] - m); sum += e[q]; }
  const float inv = 1.0f / sum;
  #pragma unroll
  for (int q = 0; q < KK; ++q) filt_out[b * CKK + base + q] = e[q] * inv;
}

// ---------------------------------------------------------------------------
// Kernel 2: transpose Wc[o][c] -> WcT[c][o] so WMMA A-fragment loads are
// coalesced along the out-channel (lane) dimension.
// ---------------------------------------------------------------------------
__global__ __launch_bounds__(256) void transpose_wc(
    const float* __restrict__ Wc, float* __restrict__ WcT) {
  const int idx = blockIdx.x * 256 + threadIdx.x;  // 65536 total
  const int o = idx & (C - 1);
  const int c = idx >> 8;
  WcT[c * C + o] = Wc[o * C + c];
}

// ---------------------------------------------------------------------------
// Kernel 3: fused depthwise 3x3 (per-sample softmax taps) -> LDS, then
// 1x1 channel-combine GEMM on v_wmma_f32_16x16x4_f32.
// One block per (b, h) image row: 256 threads = 8 waves.
//   Phase 1: LDS tmp[c][w] = depthwise result, c=0..255, w=0..63 (64 KB).
//   Phase 2: wave wv computes out channels [wv*32, wv*32+32) x all 64 px
//            as 2(M) x 4(N) tiles of 16x16, K=256 in steps of 4.
// ---------------------------------------------------------------------------
__global__ __launch_bounds__(256) void adjconv_fused(
    const float* __restrict__ img, const float* __restrict__ filt,
    const float* __restrict__ WcT, const float* __restrict__ bc,
    float* __restrict__ out) {
  __shared__ float tmp[C * W];  // 64 KB: tmp[c*64 + w]

  const int h = blockIdx.x;
  const int b = blockIdx.y;
  const int tid = threadIdx.x;

  // ---- Phase 1: depthwise sliding-window weighted sum for this row ----
  {
    const int w = tid & (W - 1);
    for (int c = (tid >> 6); c < C; c += 4) {
      const float* fl = filt + (b * C + c) * KK;
      float f[KK];
      #pragma unroll
      for (int q = 0; q < KK; ++q) f[q] = fl[q];

      const float* chan = img + ((b * C + c) * H) * W;
      float acc = 0.f;
      #pragma unroll
      for (int i = 0; i < 3; ++i) {
        const int hh = h + i - 1;
        if ((unsigned)hh < (unsigned)H) {
          const float* row = chan + hh * W;
          #pragma unroll
          for (int j = 0; j < 3; ++j) {
            const int ww = w + j - 1;
            if ((unsigned)ww < (unsigned)W) acc += f[i * 3 + j] * row[ww];
          }
        }
      }
      tmp[c * W + w] = acc;
    }
  }
  __syncthreads();

  // ---- Phase 2: 1x1 conv as GEMM: out[o,p] = sum_c WcT[c][o]*tmp[c][p] ----
  const int lane = tid & 31;
  const int wave = tid >> 5;
  const int m_base = wave * 32;         // first out-channel of this wave
  const int lm = lane & 15;             // M (A) / N (B,C,D) index within tile
  const int koff = (lane < 16) ? 0 : 2; // K offset for this lane half

  v8f acc[2][4];
  #pragma unroll
  for (int mi = 0; mi < 2; ++mi)
    #pragma unroll
    for (int ni = 0; ni < 4; ++ni)
      acc[mi][ni] = (v8f){0.f, 0.f, 0.f, 0.f, 0.f, 0.f, 0.f, 0.f};

  for (int k = 0; k < C; k += 4) {
    // A fragments: 16x4 tile of Wc^T. ISA layout (32-bit A 16x4):
    //   VGPR0: K=k+0 (lanes 0-15) / K=k+2 (lanes 16-31); VGPR1: K=k+1 / k+3.
    const float* wrow = WcT + (k + koff) * C;
    v2f a0, a1;
    a0.x = wrow[m_base + lm];
    a0.y = wrow[C + m_base + lm];
    a1.x = wrow[m_base + 16 + lm];
    a1.y = wrow[C + m_base + 16 + lm];

    // B fragments: 4x16 tiles of tmp (K x N, lane = N).
    const float* srow = tmp + (k + koff) * W;
    v2f bb[4];
    #pragma unroll
    for (int ni = 0; ni < 4; ++ni) {
      bb[ni].x = srow[ni * 16 + lm];
      bb[ni].y = srow[W + ni * 16 + lm];
    }

    #pragma unroll
    for (int ni = 0; ni < 4; ++ni) {
      acc[0][ni] = __builtin_amdgcn_wmma_f32_16x16x4_f32(
          false, a0, false, bb[ni], (short)0, acc[0][ni], false, false);
      acc[1][ni] = __builtin_amdgcn_wmma_f32_16x16x4_f32(
          false, a1, false, bb[ni], (short)0, acc[1][ni], false, false);
    }
  }

  // ---- Epilogue: D layout -> global. VGPR r: lanes 0-15 = (M=r, N=lane),
  //      lanes 16-31 = (M=r+8, N=lane-16). Add bias bc[o]. ----
  const int r_off = (lane < 16) ? 0 : 8;
  #pragma unroll
  for (int mi = 0; mi < 2; ++mi) {
    #pragma unroll
    for (int ni = 0; ni < 4; ++ni) {
      #pragma unroll
      for (int r = 0; r < 8; ++r) {
        const int o = m_base + mi * 16 + r + r_off;
        const int w = ni * 16 + lm;
        out[((b * C + o) * H + h) * W + w] = acc[mi][ni][r] + bc[o];
      }
    }
  }
}

// ---------------------------------------------------------------------------
extern "C" void kernel_launch(void* const* d_in, const int* in_sizes, int n_in,
                              void* d_out, int out_size, void* d_ws, size_t ws_size,
                              hipStream_t stream) {
  (void)in_sizes; (void)n_in; (void)out_size; (void)ws_size;
  const float* image_feat = (const float*)d_in[0];
  const float* temp_feat  = (const float*)d_in[1];
  const float* Wt = (const float*)d_in[2];
  const float* bt = (const float*)d_in[3];
  const float* Wf = (const float*)d_in[4];
  const float* bf = (const float*)d_in[5];
  const float* Wc = (const float*)d_in[6];
  const float* bc = (const float*)d_in[7];
  float* outp = (float*)d_out;

  float* ws   = (float*)d_ws;
  float* wcT  = ws;               // 256*256 floats = 256 KB
  float* filt = ws + C * C;       // 8*2304 floats  =  72 KB

  gen_filters<<<BS, 256, 0, stream>>>(temp_feat, Wt, bt, Wf, bf, filt);
  transpose_wc<<<C, 256, 0, stream>>>(Wc, wcT);
  adjconv_fused<<<dim3(H, BS), 256, 0, stream>>>(image_feat, filt, wcT, bc, outp);
}